// HeteroGCN_75505525064540
// MI455X (gfx1250) — compile-verified
//
#include <hip/hip_runtime.h>

#define N_NODES     100000
#define N_EDGES     3200000
#define IN_FEAT     32
#define H_FEAT      16
#define NUM_CLASSES 2
#define NUM_GRAPHS  512

typedef __attribute__((ext_vector_type(2))) float v2f;
typedef __attribute__((ext_vector_type(8))) float v8f;

// ---------------------------------------------------------------------------
// Fused node GEMM: for a 16-row node tile, compute
//   out_rel  = act(X) @ Wrel                  (scatter source, to be gathered per edge)
//   out_root = act(X) @ Wroot + bias          (accumulator init; edges atomically add on top)
// act = leaky_relu(0.01) if RELU (fuses previous layer's activation into the load).
// One wave per 16-row tile, V_WMMA_F32_16X16X4_F32, FIN/4 K-steps per weight matrix.
// ---------------------------------------------------------------------------
template <int FIN, bool RELU>
__global__ __launch_bounds__(256) void node_gemm_kernel(
    const float* __restrict__ X,
    const float* __restrict__ Wrel,   // [FIN][16] row-major
    const float* __restrict__ Wroot,  // [FIN][16] row-major
    const float* __restrict__ bias,   // [16]
    float* __restrict__ out_rel,      // [N_NODES][16]
    float* __restrict__ out_root,     // [N_NODES][16]
    int n_tiles) {
  const int wave = blockIdx.x * (blockDim.x >> 5) + (threadIdx.x >> 5);
  if (wave >= n_tiles) return;  // wave-uniform: EXEC stays all-ones for live waves

  const int lane = threadIdx.x & 31;
  const int r    = lane & 15;   // A row within tile / B,C column
  const int half = lane >> 4;   // K-pair selector for A/B, M-half for C/D

  constexpr int NF = FIN / 4;

  // A fragments: lane holds X[tile*16 + r][k], X[...][k+1] with k = 4f + 2*half
  v2f a[NF];
  const float* xrow = X + (size_t)(wave * 16 + r) * FIN;
#pragma unroll
  for (int f = 0; f < NF; ++f) {
    const int k = 4 * f + 2 * half;
    v2f av = *(const v2f*)(xrow + k);
    if (RELU) {
      av.x = av.x > 0.0f ? av.x : 0.01f * av.x;
      av.y = av.y > 0.0f ? av.y : 0.01f * av.y;
    }
    a[f] = av;
  }

  // B fragments: lane holds W[k][r], W[k+1][r] with k = 4f + 2*half
  v2f brel[NF], broot[NF];
#pragma unroll
  for (int f = 0; f < NF; ++f) {
    const int k = 4 * f + 2 * half;
    brel[f].x  = Wrel[k * 16 + r];
    brel[f].y  = Wrel[(k + 1) * 16 + r];
    broot[f].x = Wroot[k * 16 + r];
    broot[f].y = Wroot[(k + 1) * 16 + r];
  }

  const float bn = bias[r];  // bias per output column N = r
  v8f crel  = {};
  v8f croot;
#pragma unroll
  for (int v = 0; v < 8; ++v) croot[v] = bn;

#pragma unroll
  for (int f = 0; f < NF; ++f) {
    crel = __builtin_amdgcn_wmma_f32_16x16x4_f32(
        false, a[f], false, brel[f], (short)0, crel, false, false);
    croot = __builtin_amdgcn_wmma_f32_16x16x4_f32(
        false, a[f], false, broot[f], (short)0, croot, false, false);
  }

  // C/D layout: VGPR v, lane -> (M = v + 8*half, N = r)
#pragma unroll
  for (int v = 0; v < 8; ++v) {
    const int M = v + 8 * half;
    const size_t idx = (size_t)(wave * 16 + M) * H_FEAT + r;
    out_rel[idx]  = crel[v];
    out_root[idx] = croot[v];
  }
}

// ---------------------------------------------------------------------------
// Edge scatter: acc[dst[e]][f] += xr[src[e]][f]. 16 threads per edge (f = lane%16),
// so the gather is a coalesced 64B row read and atomics hit coalesced segments.
// Node arrays (6.4 MB) are L2-resident -> atomics resolve in L2.
// ---------------------------------------------------------------------------
__global__ __launch_bounds__(256) void edge_scatter_kernel(
    const int* __restrict__ src, const int* __restrict__ dst,
    const float* __restrict__ xr, float* __restrict__ acc) {
  const unsigned t = blockIdx.x * blockDim.x + threadIdx.x;
  const unsigned e = t >> 4;
  const unsigned f = t & 15;
  if (e >= (unsigned)N_EDGES) return;
  const int s = src[e];
  const int d = dst[e];
  atomicAdd(&acc[(size_t)d * H_FEAT + f], xr[(size_t)s * H_FEAT + f]);
}

__global__ __launch_bounds__(256) void zero_kernel(float* __restrict__ p, int n) {
  const int t = blockIdx.x * blockDim.x + threadIdx.x;
  if (t < n) p[t] = 0.0f;
}

// Graph pooling: pooled[batch[node]][f] += h[node][f]
__global__ __launch_bounds__(256) void pool_kernel(
    const float* __restrict__ h, const int* __restrict__ batch,
    float* __restrict__ pooled) {
  const unsigned t = blockIdx.x * blockDim.x + threadIdx.x;
  const unsigned node = t >> 4;
  const unsigned f = t & 15;
  if (node >= (unsigned)N_NODES) return;
  atomicAdd(&pooled[(size_t)batch[node] * H_FEAT + f], h[(size_t)node * H_FEAT + f]);
}

// Classifier: out[g][c] = pooled[g] . W_clf[:,c] + b_clf[c]   (512x2 output)
__global__ __launch_bounds__(256) void clf_kernel(
    const float* __restrict__ pooled, const float* __restrict__ Wclf,
    const float* __restrict__ bclf, float* __restrict__ out) {
  const int t = blockIdx.x * blockDim.x + threadIdx.x;
  if (t >= NUM_GRAPHS * NUM_CLASSES) return;
  const int g = t >> 1;
  const int c = t & 1;
  float s = bclf[c];
#pragma unroll
  for (int f = 0; f < H_FEAT; ++f)
    s += pooled[g * H_FEAT + f] * Wclf[f * NUM_CLASSES + c];
  out[t] = s;
}

extern "C" void kernel_launch(void* const* d_in, const int* in_sizes, int n_in,
                              void* d_out, int out_size, void* d_ws, size_t ws_size,
                              hipStream_t stream) {
  const float* x      = (const float*)d_in[0];
  const int*   ei     = (const int*)d_in[1];   // [2][N_EDGES]
  const int*   batch  = (const int*)d_in[2];
  const float* Wrel1  = (const float*)d_in[3];
  const float* Wroot1 = (const float*)d_in[4];
  const float* b1     = (const float*)d_in[5];
  const float* Wrel2  = (const float*)d_in[6];
  const float* Wroot2 = (const float*)d_in[7];
  const float* b2     = (const float*)d_in[8];
  const float* Wrel3  = (const float*)d_in[9];
  const float* Wroot3 = (const float*)d_in[10];
  const float* b3     = (const float*)d_in[11];
  const float* Wclf   = (const float*)d_in[12];
  const float* bclf   = (const float*)d_in[13];
  float* out = (float*)d_out;

  const int* src = ei;
  const int* dst = ei + N_EDGES;

  // Workspace layout (floats): xr | hA | hB | pooled  ~= 19.2 MB
  float* xr     = (float*)d_ws;
  float* hA     = xr + (size_t)N_NODES * H_FEAT;
  float* hB     = hA + (size_t)N_NODES * H_FEAT;
  float* pooled = hB + (size_t)N_NODES * H_FEAT;

  const int n_tiles     = N_NODES / 16;               // 6250 (exact)
  const int gemm_blocks = (n_tiles + 7) / 8;          // 8 waves (one tile each) per 256-thread block
  const int edge_blocks = (N_EDGES * 16 + 255) / 256; // 16 threads/edge
  const int pool_blocks = (N_NODES * 16 + 255) / 256;

  // Layer 1: FIN=32, no input activation
  node_gemm_kernel<32, false><<<gemm_blocks, 256, 0, stream>>>(
      x, Wrel1, Wroot1, b1, xr, hA, n_tiles);
  edge_scatter_kernel<<<edge_blocks, 256, 0, stream>>>(src, dst, xr, hA);

  // Layer 2: FIN=16, leaky_relu fused into A-tile load
  node_gemm_kernel<16, true><<<gemm_blocks, 256, 0, stream>>>(
      hA, Wrel2, Wroot2, b2, xr, hB, n_tiles);
  edge_scatter_kernel<<<edge_blocks, 256, 0, stream>>>(src, dst, xr, hB);

  // Layer 3: FIN=16, leaky_relu fused; no output activation
  node_gemm_kernel<16, true><<<gemm_blocks, 256, 0, stream>>>(
      hB, Wrel3, Wroot3, b3, xr, hA, n_tiles);
  edge_scatter_kernel<<<edge_blocks, 256, 0, stream>>>(src, dst, xr, hA);

  // Pool + classify
  zero_kernel<<<(NUM_GRAPHS * H_FEAT + 255) / 256, 256, 0, stream>>>(
      pooled, NUM_GRAPHS * H_FEAT);
  pool_kernel<<<pool_blocks, 256, 0, stream>>>(hA, batch, pooled);
  clf_kernel<<<(NUM_GRAPHS * NUM_CLASSES + 255) / 256, 256, 0, stream>>>(
      pooled, Wclf, bclf, out);
}